// MechanismTransformer_29205777613497
// MI455X (gfx1250) — compile-verified
//
#include <hip/hip_runtime.h>
#include <math.h>
#include <stdint.h>

#define DIMD   256
#define HEADS  8
#define DHEAD  32
#define MLPD   1024
#define ACTD   16
#define BB     8
#define SS     16
#define TT     64
#define NN     1024
#define NNP    1028            /* padded score-row stride (floats): banks = lane*4 */
#define ROWS   (BB*NN)         /* 8192 */
#define QKSCALE 0.17677669529663687f  /* 32^-0.5 */

#define KC     32              /* GEMM K-chunk staged in LDS */
#define SA_STR 36              /* padded A-tile row stride (floats) */
#define SW_STR 72              /* padded W-tile row stride (floats) */
#define TILE_F 2304            /* 64*36 == 32*72 floats per buffer */

typedef float v2f __attribute__((ext_vector_type(2)));
typedef float v8f __attribute__((ext_vector_type(8)));

__device__ __forceinline__ v8f wmma4(v2f a, v2f b, v8f c) {
  // D = A(16x4,f32) x B(4x16,f32) + C(16x16,f32) on the CDNA5 WMMA pipe
  return __builtin_amdgcn_wmma_f32_16x16x4_f32(false, a, false, b, (short)0, c,
                                               false, false);
}

// Async global -> LDS copy (VFLAT GLOBAL_LOAD_ASYNC_TO_LDS_B128, ASYNCcnt).
// ldsByte = LDS byte address (low 32 bits of generic shared pointer).
__device__ __forceinline__ void async_b128(uint32_t ldsByte, const float* g) {
  asm volatile("global_load_async_to_lds_b128 %0, %1, off"
               :: "v"(ldsByte), "v"(g)
               : "memory");
}
__device__ __forceinline__ void wait_async0() {
  asm volatile("s_wait_asynccnt 0" ::: "memory");
}
__device__ __forceinline__ uint32_t lds_byte(const void* p) {
  return (uint32_t)(uintptr_t)p;  // LDS aperture: addr[31:0] == LDS offset
}

__device__ __forceinline__ float gelu_exact(float x) {
  return 0.5f * x * (1.0f + erff(x * 0.70710678118654752f));
}
__device__ __forceinline__ float sigmoidf_(float x) { return 1.0f / (1.0f + expf(-x)); }

// ---------------------------------------------------------------------------
// WMMA fp32 GEMM: C[M,N] = epi(A[M,K] @ W[K,N]).
// Block = 256 threads = 8 waves as 4(M) x 2(N); each wave owns a 16x32 C tile
// (two accumulators reusing each A fragment). K is consumed in KC=32 chunks,
// double-buffered in LDS via async global->LDS b128 copies.
// EPI: 0 none, 1 +bias, 2 +bias,gelu, 3 +bias,+residual(R)
// Requires M%64==0, N%64==0, K%32==0 (true for all call sites here).
// ---------------------------------------------------------------------------
template <int EPI>
__global__ __launch_bounds__(256) void gemm_wmma(
    const float* __restrict__ A, const float* __restrict__ W,
    const float* __restrict__ bias, const float* __restrict__ R,
    float* __restrict__ C, int M, int K, int N) {
  __shared__ __align__(16) float sAf[2][TILE_F];  // [64][SA_STR] per buffer
  __shared__ __align__(16) float sWf[2][TILE_F];  // [KC][SW_STR] per buffer

  const int tid  = threadIdx.x;
  const int wave = tid >> 5;
  const int lane = tid & 31;
  const int wm   = wave & 3;   // M sub-tile 0..3
  const int wn   = wave >> 2;  // N half 0..1 (32 cols each)
  const int m0b  = blockIdx.y * 64;
  const int n0b  = blockIdx.x * 64;
  const int l15  = lane & 15;
  const int kh   = (lane >> 4) * 2;

  const uint32_t aBase = lds_byte(&sAf[0][0]);
  const uint32_t wBase = lds_byte(&sWf[0][0]);

  // Stage one KC-chunk (A: 64xKC, W: KCx64) into LDS buffer `buf`.
  auto stage = [&](int kc0, int buf) {
    // A: 64 rows x 8 b128 segments = 512 segments
    for (int s = tid; s < 512; s += 256) {
      const int row = s >> 3, kseg = s & 7;
      async_b128(aBase + (uint32_t)(buf * TILE_F + row * SA_STR + kseg * 4) * 4,
                 A + (size_t)(m0b + row) * K + kc0 + kseg * 4);
    }
    // W: KC rows x 16 b128 segments = 512 segments
    for (int s = tid; s < 512; s += 256) {
      const int row = s >> 4, nseg = s & 15;
      async_b128(wBase + (uint32_t)(buf * TILE_F + row * SW_STR + nseg * 4) * 4,
                 W + (size_t)(kc0 + row) * N + n0b + nseg * 4);
    }
  };

  v8f acc0 = {}, acc1 = {};
  const int nch = K / KC;
  stage(0, 0);
  for (int c = 0; c < nch; ++c) {
    wait_async0();
    __syncthreads();              // staged chunk visible to all waves
    if (c + 1 < nch) stage((c + 1) * KC, (c + 1) & 1);
    const float* __restrict__ sA = sAf[c & 1];
    const float* __restrict__ sW = sWf[c & 1];
#pragma unroll
    for (int kk = 0; kk < KC; kk += 4) {
      v2f a, b0, b1;
      a.x  = sA[(wm * 16 + l15) * SA_STR + kk + kh];
      a.y  = sA[(wm * 16 + l15) * SA_STR + kk + kh + 1];
      b0.x = sW[(kk + kh) * SW_STR + wn * 32 + l15];
      b0.y = sW[(kk + kh + 1) * SW_STR + wn * 32 + l15];
      b1.x = sW[(kk + kh) * SW_STR + wn * 32 + 16 + l15];
      b1.y = sW[(kk + kh + 1) * SW_STR + wn * 32 + 16 + l15];
      acc0 = wmma4(a, b0, acc0);
      acc1 = wmma4(a, b1, acc1);
    }
    __syncthreads();              // all waves done before buffer is re-staged
  }

  const int mbase = m0b + wm * 16 + (lane >> 4) * 8;
  const int n0 = n0b + wn * 32 + l15;
  float bv0 = 0.f, bv1 = 0.f;
  if (EPI >= 1) { bv0 = bias[n0]; bv1 = bias[n0 + 16]; }
#pragma unroll
  for (int r = 0; r < 8; ++r) {
    const size_t i0 = (size_t)(mbase + r) * N + n0;
    const size_t i1 = i0 + 16;
    float v0 = acc0[r], v1 = acc1[r];
    if (EPI >= 1) { v0 += bv0; v1 += bv1; }
    if (EPI == 2) { v0 = gelu_exact(v0); v1 = gelu_exact(v1); }
    if (EPI == 3) { v0 += R[i0]; v1 += R[i1]; }
    C[i0] = v0;
    C[i1] = v1;
  }
}

// ---------------------------------------------------------------------------
// Fused attention for one (b, h, 16-row i-tile).
// qkv row layout: [q(256) | k(256) | v(256)], head h uses cols h*32..h*32+31.
// Dynamic LDS: scores[16][NNP] + obuf[2][16][16].
// ---------------------------------------------------------------------------
__global__ __launch_bounds__(256) void attn_kernel(
    const float* __restrict__ qkv, const float* __restrict__ gate,
    float* __restrict__ o) {
  extern __shared__ float smem[];
  float* sS = smem;             // [16][NNP]
  float* sO = smem + 16 * NNP;  // [2][16][16]

  const int tid  = threadIdx.x;
  const int wave = tid >> 5;
  const int lane = tid & 31;
  const int i0 = blockIdx.x * 16;
  const int h  = blockIdx.y;
  const int b  = blockIdx.z;

  const float* __restrict__ base = qkv + (size_t)b * NN * 768;
  const float* __restrict__ Qp = base + h * DHEAD;
  const float* __restrict__ Kp = base + DIMD + h * DHEAD;
  const float* __restrict__ Vp = base + 2 * DIMD + h * DHEAD;

  const int kh  = (lane >> 4) * 2;
  const int l15 = lane & 15;

  // Q tile fragments (16x32), pre-scaled
  v2f qa[8];
  {
    const float* Qrow = Qp + (size_t)(i0 + l15) * 768;
#pragma unroll
    for (int kk = 0; kk < 8; ++kk) {
      qa[kk].x = Qrow[4 * kk + kh] * QKSCALE;
      qa[kk].y = Qrow[4 * kk + kh + 1] * QKSCALE;
    }
  }

  // Phase 1: S = (Q*scale) @ K^T ; each wave owns 8 j-tiles of 16
  for (int it = 0; it < 8; ++it) {
    const int j0 = (wave + 8 * it) * 16;
    const float* Krow = Kp + (size_t)(j0 + l15) * 768;
    v8f acc = {};
#pragma unroll
    for (int kk = 0; kk < 8; ++kk) {
      v2f bf;
      bf.x = Krow[4 * kk + kh];      // B[k][n] = K[j0+n][k]
      bf.y = Krow[4 * kk + kh + 1];
      acc = wmma4(qa[kk], bf, acc);
    }
    const int mb = (lane >> 4) * 8;
#pragma unroll
    for (int r = 0; r < 8; ++r) sS[(mb + r) * NNP + j0 + l15] = acc[r];
  }
  sO[tid] = 0.0f;
  sO[tid + 256] = 0.0f;
  __syncthreads();

  // Phase 2: exact softmax over j (1024) + mechanism gate (post-softmax)
  {
    const int row = tid >> 4;  // 16 threads (one half-wave) per row
    const int sub = tid & 15;
    float* Sr = sS + row * NNP;
    float mx = -1e30f;
    for (int j = sub; j < NN; j += 16) mx = fmaxf(mx, Sr[j]);
    for (int off = 8; off; off >>= 1) mx = fmaxf(mx, __shfl_xor(mx, off, 16));
    float sum = 0.0f;
    for (int j = sub; j < NN; j += 16) {
      float e = expf(Sr[j] - mx);
      Sr[j] = e;
      sum += e;
    }
    for (int off = 8; off; off >>= 1) sum += __shfl_xor(sum, off, 16);
    const float inv = 1.0f / sum;
    // s_i = (i0+row)%16 = row (i0 multiple of 16); s_j = j%16
    const float* gp = gate + (((size_t)b * HEADS + h) * SS + row) * SS;
    for (int j = sub; j < NN; j += 16) Sr[j] = Sr[j] * inv * gp[j & 15];
  }
  __syncthreads();

  // Phase 3: O(16x32) = P(16x1024) @ V(1024x32); K split across 4 wave-pairs
  {
    const int n0 = (wave & 1) * 16;
    const int kb = (wave >> 1) * 256;
    v8f acc = {};
    for (int k0 = kb; k0 < kb + 256; k0 += 4) {
      v2f a, bf;
      a.x = sS[l15 * NNP + k0 + kh];
      a.y = sS[l15 * NNP + k0 + kh + 1];
      bf.x = Vp[(size_t)(k0 + kh) * 768 + n0 + l15];
      bf.y = Vp[(size_t)(k0 + kh + 1) * 768 + n0 + l15];
      acc = wmma4(a, bf, acc);
    }
    const int mb = (lane >> 4) * 8;
    float* dst = sO + (wave & 1) * 256;
#pragma unroll
    for (int r = 0; r < 8; ++r) atomicAdd(&dst[(mb + r) * 16 + l15], acc[r]);
  }
  __syncthreads();

  for (int idx = tid; idx < 512; idx += 256) {
    const int tile = idx >> 8;
    const int m  = (idx >> 4) & 15;
    const int nl = idx & 15;
    o[((size_t)b * NN + i0 + m) * DIMD + h * DHEAD + tile * 16 + nl] = sO[idx];
  }
}

// ---------------------------------------------------------------------------
// Small VALU kernels
// ---------------------------------------------------------------------------
__global__ __launch_bounds__(256) void prep_action_kernel(
    const float* __restrict__ actions, const float* __restrict__ w1,
    const float* __restrict__ b1, const float* __restrict__ w2,
    const float* __restrict__ b2, float* __restrict__ sae,
    float* __restrict__ sae2) {
  __shared__ float hid[128];
  const int bt = blockIdx.x;  // b*TT + t
  const int tid = threadIdx.x;
  if (tid < 128) {
    float s = b1[tid];
    const float* act = actions + bt * ACTD;
#pragma unroll
    for (int k = 0; k < ACTD; ++k) s += act[k] * w1[k * 128 + tid];
    hid[tid] = gelu_exact(s);
  }
  __syncthreads();
  float s = b2[tid];
  for (int j = 0; j < 128; ++j) s += hid[j] * w2[j * DIMD + tid];
  const float si = s * sigmoidf_(s);  // silu(action_emb)
  sae[(size_t)bt * DIMD + tid] = si;
  sae2[(size_t)bt * 2 * DIMD + tid] = si;
  sae2[(size_t)bt * 2 * DIMD + DIMD + tid] = si;
}

__global__ __launch_bounds__(256) void mech_context_kernel(
    const float* __restrict__ m_ij, float* __restrict__ mc) {
  const int bs = blockIdx.x;  // b*SS + s
  const int tid = threadIdx.x;
  float s = 0.f;
  const float* p = m_ij + (size_t)bs * SS * DIMD + tid;
  for (int j = 0; j < SS; ++j) s += p[j * DIMD];
  mc[(size_t)bs * DIMD + tid] = s * (1.0f / SS);
}

__global__ __launch_bounds__(256) void gate_kernel(
    const float* __restrict__ m_ij, const float* __restrict__ w1,
    const float* __restrict__ b1, const float* __restrict__ w2,
    const float* __restrict__ b2, float* __restrict__ g) {
  __shared__ float hid[128];
  const int rid = blockIdx.x;  // (b*SS+si)*SS + sj
  const int tid = threadIdx.x;
  const float* mrow = m_ij + (size_t)rid * DIMD;
  if (tid < 128) {
    float s = b1[tid];
    for (int k = 0; k < DIMD; ++k) s += mrow[k] * w1[k * 128 + tid];
    hid[tid] = gelu_exact(s);
  }
  __syncthreads();
  if (tid < HEADS) {
    float s = b2[tid];
    for (int j = 0; j < 128; ++j) s += hid[j] * w2[j * HEADS + tid];
    const int sj = rid & 15, si = (rid >> 4) & 15, b = rid >> 8;
    g[(((size_t)b * HEADS + tid) * SS + si) * SS + sj] = sigmoidf_(s);
  }
}

__device__ __forceinline__ float block_sum_256(float v, float* red, int tid) {
  red[tid] = v;
  __syncthreads();
  for (int s = 128; s > 0; s >>= 1) {
    if (tid < s) red[tid] += red[tid + s];
    __syncthreads();
  }
  float r = red[0];
  __syncthreads();
  return r;
}

__global__ __launch_bounds__(256) void ln_mod_kernel(
    const float* __restrict__ x, const float* __restrict__ mod,
    const float* __restrict__ lng, const float* __restrict__ lnb,
    float* __restrict__ xn) {
  __shared__ float red[256];
  const int row = blockIdx.x;  // b*NN + n
  const int tid = threadIdx.x;
  const float v = x[(size_t)row * DIMD + tid];
  const float mean = block_sum_256(v, red, tid) * (1.0f / DIMD);
  const float d = v - mean;
  const float var = block_sum_256(d * d, red, tid) * (1.0f / DIMD);
  float xv = d * rsqrtf(var + 1e-5f) * lng[tid] + lnb[tid];
  const int b = row >> 10;
  const int t = (row & (NN - 1)) >> 4;
  const float* mp = mod + ((size_t)b * TT + t) * (2 * DIMD);
  xn[(size_t)row * DIMD + tid] = xv * (1.0f + mp[tid]) + mp[DIMD + tid];
}

__global__ __launch_bounds__(256) void ffn_ln_mod_kernel(
    const float* __restrict__ x, const float* __restrict__ mc,
    const float* __restrict__ fmod, const float* __restrict__ lng,
    const float* __restrict__ lnb, float* __restrict__ comb) {
  __shared__ float red[256];
  const int row = blockIdx.x;
  const int tid = threadIdx.x;
  const int b = row >> 10;
  const int n = row & (NN - 1);
  const int t = n >> 4;
  const int sl = n & 15;
  const float e0 = x[(size_t)row * DIMD + tid];
  const float e1 = mc[((size_t)b * SS + sl) * DIMD + tid];
  const float mean = block_sum_256(e0 + e1, red, tid) * (1.0f / (2 * DIMD));
  const float d0 = e0 - mean, d1 = e1 - mean;
  const float var = block_sum_256(d0 * d0 + d1 * d1, red, tid) * (1.0f / (2 * DIMD));
  const float inv = rsqrtf(var + 1e-5f);
  const float* fp = fmod + ((size_t)b * TT + t) * (4 * DIMD);
  float c0 = (d0 * inv) * lng[tid] + lnb[tid];
  float c1 = (d1 * inv) * lng[DIMD + tid] + lnb[DIMD + tid];
  c0 = c0 * (1.0f + fp[tid]) + fp[2 * DIMD + tid];
  c1 = c1 * (1.0f + fp[DIMD + tid]) + fp[2 * DIMD + DIMD + tid];
  comb[(size_t)row * 2 * DIMD + tid] = c0;
  comb[(size_t)row * 2 * DIMD + DIMD + tid] = c1;
}

__global__ __launch_bounds__(256) void final_ln_kernel(
    const float* __restrict__ x, const float* __restrict__ g,
    const float* __restrict__ b, float* __restrict__ out) {
  __shared__ float red[256];
  const int row = blockIdx.x;
  const int tid = threadIdx.x;
  const float v = x[(size_t)row * DIMD + tid];
  const float mean = block_sum_256(v, red, tid) * (1.0f / DIMD);
  const float d = v - mean;
  const float var = block_sum_256(d * d, red, tid) * (1.0f / DIMD);
  out[(size_t)row * DIMD + tid] = d * rsqrtf(var + 1e-5f) * g[tid] + b[tid];
}

// ---------------------------------------------------------------------------
// Host launcher
// ---------------------------------------------------------------------------
extern "C" void kernel_launch(void* const* d_in, const int* in_sizes, int n_in,
                              void* d_out, int out_size, void* d_ws,
                              size_t ws_size, hipStream_t stream) {
  (void)in_sizes; (void)n_in; (void)out_size; (void)ws_size;

  // Top-level dict order: x, m_ij, actions, params (jax sorted-key leaves), num_slots
  const float* x_in   = (const float*)d_in[0];
  const float* m_ij   = (const float*)d_in[1];
  const float* acts   = (const float*)d_in[2];
  const float* act_b1 = (const float*)d_in[3];
  const float* act_b2 = (const float*)d_in[4];
  const float* act_w1 = (const float*)d_in[5];
  const float* act_w2 = (const float*)d_in[6];
  const float* fin_b  = (const float*)d_in[7];
  const float* fin_g  = (const float*)d_in[8];

  // Workspace carve-up (fp32), with buffer reuse: regA = qkv then ffn-hidden,
  // regB = xn then comb.
  float* ws = (float*)d_ws;
  size_t off = 0;
  auto alloc = [&](size_t n) { float* p = ws + off; off += n; return p; };
  float* x_buf = alloc((size_t)ROWS * DIMD);
  float* obuf  = alloc((size_t)ROWS * DIMD);
  float* regA  = alloc((size_t)ROWS * MLPD);
  float* regB  = alloc((size_t)ROWS * 2 * DIMD);
  float* sae   = alloc((size_t)BB * TT * DIMD);
  float* sae2  = alloc((size_t)BB * TT * 2 * DIMD);
  float* modb  = alloc((size_t)BB * TT * 2 * DIMD);
  float* fmodb = alloc((size_t)BB * TT * 4 * DIMD);
  float* mcb   = alloc((size_t)BB * SS * DIMD);
  float* gbuf  = alloc((size_t)BB * HEADS * SS * SS);

  hipMemcpyAsync(x_buf, x_in, sizeof(float) * (size_t)ROWS * DIMD,
                 hipMemcpyDeviceToDevice, stream);

  prep_action_kernel<<<BB * TT, 256, 0, stream>>>(acts, act_w1, act_b1, act_w2,
                                                  act_b2, sae, sae2);
  mech_context_kernel<<<BB * SS, 256, 0, stream>>>(m_ij, mcb);

  for (int L = 0; L < 2; ++L) {
    const float* const* lp = (const float* const*)(d_in + 9 + 19 * L);
    const float* adaln_b  = lp[0];
    const float* adaln_w  = lp[1];
    const float* fadaln_b = lp[2];
    const float* fadaln_w = lp[3];
    const float* ffn_b1   = lp[4];
    const float* ffn_b2   = lp[5];
    const float* ffn_w1   = lp[6];
    const float* ffn_w2   = lp[7];
    const float* fln_b    = lp[8];
    const float* fln_g    = lp[9];
    const float* gate_b1  = lp[10];
    const float* gate_b2  = lp[11];
    const float* gate_w1  = lp[12];
    const float* gate_w2  = lp[13];
    const float* ln_b     = lp[14];
    const float* ln_g     = lp[15];
    const float* out_b    = lp[16];
    const float* out_w    = lp[17];
    const float* qkv_w    = lp[18];

    // mod = silu(action_emb) @ adaln_w + adaln_b   (rows = B*T = 512)
    gemm_wmma<1><<<dim3(8, 8), 256, 0, stream>>>(sae, adaln_w, adaln_b,
                                                 nullptr, modb, BB * TT, DIMD,
                                                 2 * DIMD);
    // xn = LN(x)*(1+sc)+sh
    ln_mod_kernel<<<ROWS, 256, 0, stream>>>(x_buf, modb, ln_g, ln_b, regB);
    // qkv = xn @ qkv_w (no bias)
    gemm_wmma<0><<<dim3(12, ROWS / 64), 256, 0, stream>>>(
        regB, qkv_w, nullptr, nullptr, regA, ROWS, DIMD, 3 * DIMD);
    // mechanism gate table (B,H,16,16)
    gate_kernel<<<BB * SS * SS, 256, 0, stream>>>(m_ij, gate_w1, gate_b1,
                                                  gate_w2, gate_b2, gbuf);
    // fused attention
    attn_kernel<<<dim3(NN / 16, HEADS, BB), 256,
                  (16 * NNP + 512) * sizeof(float), stream>>>(regA, gbuf, obuf);
    // x += o @ out_w + out_b
    gemm_wmma<3><<<dim3(4, ROWS / 64), 256, 0, stream>>>(
        obuf, out_w, out_b, x_buf, x_buf, ROWS, DIMD, DIMD);
    // fmod = silu(ae2) @ fadaln_w + fadaln_b   (rows = B*T)
    gemm_wmma<1><<<dim3(16, 8), 256, 0, stream>>>(sae2, fadaln_w, fadaln_b,
                                                  nullptr, fmodb, BB * TT,
                                                  2 * DIMD, 4 * DIMD);
    // comb = LN(concat(x, mc))*(1+fsc)+fsh
    ffn_ln_mod_kernel<<<ROWS, 256, 0, stream>>>(x_buf, mcb, fmodb, fln_g,
                                                fln_b, regB);
    // h = gelu(comb @ ffn_w1 + ffn_b1)
    gemm_wmma<2><<<dim3(16, ROWS / 64), 256, 0, stream>>>(
        regB, ffn_w1, ffn_b1, nullptr, regA, ROWS, 2 * DIMD, MLPD);
    // x += h @ ffn_w2 + ffn_b2
    gemm_wmma<3><<<dim3(4, ROWS / 64), 256, 0, stream>>>(
        regA, ffn_w2, ffn_b2, x_buf, x_buf, ROWS, MLPD, DIMD);
  }

  final_ln_kernel<<<ROWS, 256, 0, stream>>>(x_buf, fin_g, fin_b,
                                            (float*)d_out);
}